// HeteroRGCNLayer_5927054869107
// MI455X (gfx1250) — compile-verified
//
#include <hip/hip_runtime.h>

typedef float v2f __attribute__((ext_vector_type(2)));
typedef float v8f __attribute__((ext_vector_type(8)));

// ---------------------------------------------------------------------------
// Zero-fill (grid-stride). d_out doubles as the segment-sum accumulator, so it
// must be cleared at the start of every launch (graph replays don't re-poison).
// ---------------------------------------------------------------------------
__global__ __launch_bounds__(256) void zero_f32(float* __restrict__ p, long long n) {
  long long i = (long long)blockIdx.x * blockDim.x + threadIdx.x;
  long long stride = (long long)gridDim.x * blockDim.x;
  for (; i < n; i += stride) p[i] = 0.0f;
}

// ---------------------------------------------------------------------------
// Wh = feat @ W.T + bias   (feat [nrows,64], W [64,64] row-major, Wh [nrows,64])
// One wave -> 16 rows x 64 cols via 4 accumulators, K looped 16x4 with
// V_WMMA_F32_16X16X4_F32 (full fp32, matches reference numerics).
//
// A 16x4 f32 frag: lane&15 = M; half=lane>>4: VGPR{0,1} = K {k0,k0+1} for
// half 0, {k0+2,k0+3} for half 1. B 4x16 frag mirrors with lane&15 = N.
// C/D: VGPR v -> M = v + 8*half, N = lane&15.
// ---------------------------------------------------------------------------
__global__ __launch_bounds__(256) void gemm64_wmma(
    const float* __restrict__ feat, const float* __restrict__ W,
    const float* __restrict__ bias, float* __restrict__ Wh, int nrows)
{
  const int lane = threadIdx.x & 31;
  const int wave = threadIdx.x >> 5;
  const int row0 = (blockIdx.x * 8 + wave) * 16;
  if (row0 >= nrows) return;                 // wave-granular: EXEC stays all-1s

  const int half = lane >> 4;                // 0 or 1
  const int l15  = lane & 15;

  // init accumulators with bias (depends only on column N = l15)
  v8f acc[4];
#pragma unroll
  for (int o = 0; o < 4; ++o) {
    const float bv = bias[o * 16 + l15];
#pragma unroll
    for (int i = 0; i < 8; ++i) acc[o][i] = bv;
  }

  // clamp A-row for (hypothetical) partial tiles; garbage rows never stored
  int arow_idx = row0 + l15;
  if (arow_idx >= nrows) arow_idx = nrows - 1;
  const float* __restrict__ arow = feat + (long long)arow_idx * 64;

#pragma unroll
  for (int ks = 0; ks < 16; ++ks) {
    const int k0 = ks * 4 + half * 2;        // even -> 8B-aligned float2 loads
    const v2f a = *reinterpret_cast<const v2f*>(arow + k0);
#pragma unroll
    for (int o = 0; o < 4; ++o) {
      // B[k][n] = W[n][k]; lane holds column n = o*16 + l15
      const v2f b = *reinterpret_cast<const v2f*>(
          W + ((long long)(o * 16 + l15)) * 64 + k0);
      acc[o] = __builtin_amdgcn_wmma_f32_16x16x4_f32(
          false, a, false, b, (short)0, acc[o], false, false);
    }
  }

  // store: element v of acc[o] -> row row0 + v + 8*half, col o*16 + l15
#pragma unroll
  for (int o = 0; o < 4; ++o) {
#pragma unroll
    for (int v = 0; v < 8; ++v) {
      const int r = row0 + v + half * 8;
      if (r < nrows) Wh[(long long)r * 64 + o * 16 + l15] = acc[o][v];
    }
  }
}

// ---------------------------------------------------------------------------
// Segment-sum scatter: one wave32 per edge; lane handles 2 consecutive floats
// (coalesced 256B/wave gather, f32 atomics land in the 192MB L2 since msum is
// only 25.6MB). Lane 0 accumulates the destination degree.
// ---------------------------------------------------------------------------
__global__ __launch_bounds__(256) void scatter_mean_accum(
    const float* __restrict__ Wh, const int* __restrict__ src,
    const int* __restrict__ dst, float* __restrict__ msum,
    float* __restrict__ deg, int nedges)
{
  const int gid = blockIdx.x * 256 + threadIdx.x;
  const int e = gid >> 5;
  if (e >= nedges) return;
  const int lane = threadIdx.x & 31;

  const int s = src[e];
  const int d = dst[e];

  const v2f v = *reinterpret_cast<const v2f*>(Wh + (long long)s * 64 + lane * 2);
  float* out = msum + (long long)d * 64 + lane * 2;
  atomicAdd(out,     v.x);
  atomicAdd(out + 1, v.y);
  if (lane == 0) atomicAdd(deg + d, 1.0f);
}

// ---------------------------------------------------------------------------
// In-place mean: h[n,:] /= max(deg[n], 1)
// ---------------------------------------------------------------------------
__global__ __launch_bounds__(256) void normalize_mean(
    float* __restrict__ h, const float* __restrict__ deg, int nnodes)
{
  const long long gid = (long long)blockIdx.x * 256 + threadIdx.x;
  if (gid >= (long long)nnodes * 64) return;
  const float dg = deg[(int)(gid >> 6)];
  h[gid] = h[gid] / fmaxf(dg, 1.0f);
}

// ---------------------------------------------------------------------------
extern "C" void kernel_launch(void* const* d_in, const int* in_sizes, int n_in,
                              void* d_out, int out_size, void* d_ws, size_t ws_size,
                              hipStream_t stream)
{
  const float* feat_user = (const float*)d_in[0];
  const float* feat_item = (const float*)d_in[1];
  const int*   src_clicks = (const int*)d_in[2];
  const int*   dst_clicks = (const int*)d_in[3];
  const int*   src_cb     = (const int*)d_in[4];
  const int*   dst_cb     = (const int*)d_in[5];
  const float* W_clicks   = (const float*)d_in[6];
  const float* b_clicks   = (const float*)d_in[7];
  const float* W_cb       = (const float*)d_in[8];
  const float* b_cb       = (const float*)d_in[9];

  const int n_user  = in_sizes[0] / 64;
  const int n_item  = in_sizes[1] / 64;
  const int n_edges = in_sizes[2];

  float* out    = (float*)d_out;
  float* h_user = out;                           // [n_user, 64]  (relation clicked_by)
  float* h_item = out + (long long)n_user * 64;  // [n_item, 64]  (relation clicks)

  // workspace: Wh_clicks | Wh_cb | deg_item | deg_user
  float* ws        = (float*)d_ws;
  float* Wh_clicks = ws;                                   // n_user*64
  float* Wh_cb     = Wh_clicks + (long long)n_user * 64;   // n_item*64
  float* deg_item  = Wh_cb + (long long)n_item * 64;       // n_item
  float* deg_user  = deg_item + n_item;                    // n_user

  // 1) clear accumulators (output buffer) + degree counters (contiguous)
  zero_f32<<<2048, 256, 0, stream>>>(out, (long long)(n_user + n_item) * 64);
  zero_f32<<<256, 256, 0, stream>>>(deg_item, (long long)(n_item + n_user));

  // 2) per-relation dense linear via fp32 WMMA
  const int gb_user = (n_user + 127) / 128;
  const int gb_item = (n_item + 127) / 128;
  gemm64_wmma<<<gb_user, 256, 0, stream>>>(feat_user, W_clicks, b_clicks, Wh_clicks, n_user);
  gemm64_wmma<<<gb_item, 256, 0, stream>>>(feat_item, W_cb, b_cb, Wh_cb, n_item);

  // 3) edge scatter (copy_u + segment-sum + degree)
  const long long sthreads = (long long)n_edges * 32;
  const int sblocks = (int)((sthreads + 255) / 256);
  scatter_mean_accum<<<sblocks, 256, 0, stream>>>(Wh_clicks, src_clicks, dst_clicks,
                                                  h_item, deg_item, n_edges);
  scatter_mean_accum<<<sblocks, 256, 0, stream>>>(Wh_cb, src_cb, dst_cb,
                                                  h_user, deg_user, n_edges);

  // 4) mean normalization (zero-degree nodes -> 0 since accumulator is 0)
  normalize_mean<<<((long long)n_user * 64 + 255) / 256, 256, 0, stream>>>(h_user, deg_user, n_user);
  normalize_mean<<<((long long)n_item * 64 + 255) / 256, 256, 0, stream>>>(h_item, deg_item, n_item);
}